// LinearSelfAttention_37288906064390
// MI455X (gfx1250) — compile-verified
//
#include <hip/hip_runtime.h>
#include <math.h>

typedef __attribute__((ext_vector_type(2))) float v2f;
typedef __attribute__((ext_vector_type(8))) float v8f;
typedef __attribute__((ext_vector_type(4))) unsigned int u32x4;
typedef __attribute__((ext_vector_type(8))) int i32x8;
typedef __attribute__((ext_vector_type(4))) int i32x4;
typedef int v4i __attribute__((vector_size(16)));

#if __has_builtin(__builtin_amdgcn_global_load_async_to_lds_b128)
#define HAVE_ASYNC_CP 1
#else
#define HAVE_ASYNC_CP 0
#endif

#if __has_builtin(__builtin_amdgcn_tensor_load_to_lds)
#define HAVE_TDM 1
#else
#define HAVE_TDM 0
#endif

// D = A(16x4) * B(4x16) + C, fp32 WMMA (CDNA5 wave32)
__device__ inline v8f wmma4(v2f a, v2f b, v8f c) {
  return __builtin_amdgcn_wmma_f32_16x16x4_f32(
      /*neg_a=*/false, a, /*neg_b=*/false, b,
      /*c_mod=*/(short)0, c, /*reuse_a=*/false, /*reuse_b=*/false);
}

__device__ inline v8f vzero8() {
  v8f z = {0.f, 0.f, 0.f, 0.f, 0.f, 0.f, 0.f, 0.f};
  return z;
}

#if HAVE_ASYNC_CP
// 16-byte global -> LDS async copy (no VGPR round trip, tracked by ASYNCcnt)
__device__ inline void async_cp16(const float* g, float* l) {
  __builtin_amdgcn_global_load_async_to_lds_b128(
      (__attribute__((address_space(1))) v4i*)
          (const __attribute__((address_space(1))) void*)g,
      (__attribute__((address_space(3))) v4i*)
          (__attribute__((address_space(3))) void*)l,
      0, 0);
}
#endif

__device__ inline void wait_async0() {
#if __has_builtin(__builtin_amdgcn_s_wait_asynccnt)
  __builtin_amdgcn_s_wait_asynccnt(0);
#else
  asm volatile("s_wait_asynccnt 0" ::: "memory");
#endif
}

#if HAVE_TDM
__device__ inline unsigned lds_offset_u32(const void* p) {
  return (unsigned)(unsigned long long)
      (const __attribute__((address_space(3))) void*)p;
}
#endif

#define D_MODEL 1024
#define N_HEAD  16
#define D_HEAD  64
#define M_FEAT  128
#define L_SEQ   4096
#define DN_SCALE    0.3535533906f   // 1/64^0.25
#define RATIO_SCALE 0.08838834765f  // 1/sqrt(128)

// ---------------------------------------------------------------------------
// Generic fp32 GEMM: C[M,N] = A[M,K] * B[K,N]  (row-major, M%128==0, N%64==0,
// K%16==0). 256 threads = 8 waves; block tile 128x64; wave tile 32x32.
// Double-buffered LDS, async global->LDS copies when available.
// ---------------------------------------------------------------------------
#define BM 128
#define BN 64
#define BK 16
#define APAD 4  // A row stride 20 floats = 80B (16B aligned)
#define BPAD 4  // B row stride 68 floats = 272B (16B aligned)

__device__ inline void stage_tiles(const float* __restrict__ A,
                                   const float* __restrict__ B, int K, int N,
                                   int bm, int bn, int k0, int ar, int ac,
                                   int br, int bc, float (*As)[BK + APAD],
                                   float (*Bs)[BN + BPAD]) {
  const float* asrc = A + (size_t)(bm + ar) * K + k0 + ac;
  const float* bsrc = B + (size_t)(k0 + br) * N + bn + bc;
#if HAVE_ASYNC_CP
  async_cp16(asrc, &As[ar][ac]);
  async_cp16(asrc + 4, &As[ar][ac + 4]);
  async_cp16(bsrc, &Bs[br][bc]);
#else
  #pragma unroll
  for (int c = 0; c < 8; ++c) As[ar][ac + c] = asrc[c];
  #pragma unroll
  for (int c = 0; c < 4; ++c) Bs[br][bc + c] = bsrc[c];
#endif
}

__global__ __launch_bounds__(256) void gemm_f32_wmma(
    const float* __restrict__ A, const float* __restrict__ B,
    float* __restrict__ C, int M, int N, int K) {
  __shared__ float As[2][BM][BK + APAD];
  __shared__ float Bs[2][BK][BN + BPAD];

  const int tid  = threadIdx.x;
  const int lane = tid & 31;
  const int wave = tid >> 5;
  const int l15  = lane & 15;
  const int hi   = lane >> 4;

  const int bm = blockIdx.x * BM;
  const int bn = blockIdx.y * BN;
  const int wm = (wave & 3) * 32;
  const int wn = (wave >> 2) * 32;

  v8f acc[2][2];
  #pragma unroll
  for (int i = 0; i < 2; ++i)
    #pragma unroll
    for (int j = 0; j < 2; ++j) acc[i][j] = vzero8();

  const int ar = tid >> 1;          // 0..127
  const int ac = (tid & 1) * 8;     // 0 or 8
  const int br = tid >> 4;          // 0..15
  const int bc = (tid & 15) * 4;    // 0..60

  stage_tiles(A, B, K, N, bm, bn, 0, ar, ac, br, bc, As[0], Bs[0]);

  int buf = 0;
  for (int k0 = 0; k0 < K; k0 += BK, buf ^= 1) {
#if HAVE_ASYNC_CP
    wait_async0();   // my copies for tile `buf` landed
#endif
    __syncthreads(); // everyone's copies landed; prev compute done
    if (k0 + BK < K)
      stage_tiles(A, B, K, N, bm, bn, k0 + BK, ar, ac, br, bc,
                  As[buf ^ 1], Bs[buf ^ 1]);

    #pragma unroll
    for (int ks = 0; ks < BK; ks += 4) {
      v2f af[2], bf[2];
      #pragma unroll
      for (int mi = 0; mi < 2; ++mi) {
        const int row = wm + mi * 16 + l15;
        af[mi].x = As[buf][row][ks + 2 * hi];
        af[mi].y = As[buf][row][ks + 2 * hi + 1];
      }
      #pragma unroll
      for (int ni = 0; ni < 2; ++ni) {
        const int col = wn + ni * 16 + l15;
        bf[ni].x = Bs[buf][ks + 2 * hi][col];
        bf[ni].y = Bs[buf][ks + 2 * hi + 1][col];
      }
      #pragma unroll
      for (int mi = 0; mi < 2; ++mi)
        #pragma unroll
        for (int ni = 0; ni < 2; ++ni)
          acc[mi][ni] = wmma4(af[mi], bf[ni], acc[mi][ni]);
    }
  }

  #pragma unroll
  for (int mi = 0; mi < 2; ++mi)
    #pragma unroll
    for (int ni = 0; ni < 2; ++ni)
      #pragma unroll
      for (int r = 0; r < 8; ++r) {
        const int row = bm + wm + mi * 16 + r + 8 * hi;
        const int col = bn + wn + ni * 16 + l15;
        C[(size_t)row * N + col] = acc[mi][ni][r];
      }
}

// ---------------------------------------------------------------------------
// kv kernel: per (b, h, lseg): accumulate kv[b,h][256][64] over 512 tokens.
// key_pos + Fourier dash (WMMA) + sin/cos + k'^T @ v (WMMA). The 16x64 v-tile
// is DMA'd into LDS by the Tensor Data Mover, overlapped with the dash phase.
// ---------------------------------------------------------------------------
__global__ __launch_bounds__(256) void kv_kernel(
    const float* __restrict__ pos_proj, const float* __restrict__ slope_proj,
    const float* __restrict__ v, const float* __restrict__ proj,
    const float* __restrict__ scale_v, const float* __restrict__ offs_v,
    float* __restrict__ kv_g) {
  const int wg   = blockIdx.x;
  const int lseg = wg & 7;
  const int h    = (wg >> 3) & 15;
  const int b    = wg >> 7;

  __shared__ float kT[2 * M_FEAT][18];  // transposed k' tile [feat][tok]
  __shared__ float vS[16][D_HEAD];      // v tile [tok][d] (TDM destination)

  const int tid  = threadIdx.x;
  const int lane = tid & 31;
  const int wave = tid >> 5;
  const int l15  = lane & 15;
  const int hi   = lane >> 4;

  const float sc    = scale_v[h];
  const float off   = offs_v[h];
  const float a_mul = DN_SCALE * sc;
  const float b_mul = DN_SCALE * sc * off;

  v8f acc[2][4];
  #pragma unroll
  for (int i = 0; i < 2; ++i)
    #pragma unroll
    for (int j = 0; j < 4; ++j) acc[i][j] = vzero8();

#if HAVE_TDM
  // Tensor DMA descriptor: 2D tensor, element 4B, 64 wide x 16 rows,
  // row stride 1024 elements; tile == tensor extent.
  u32x4 td_g0;
  i32x8 td_g1;
  i32x4 td_g2 = {0, 0, 0, 0};
  i32x4 td_g3 = {0, 0, 0, 0};
  i32x8 td_g4 = {0, 0, 0, 0, 0, 0, 0, 0};
  td_g0[0] = 1u;                             // count=1, user descriptor
  td_g0[1] = lds_offset_u32(&vS[0][0]);      // lds_addr
  td_g1[0] = 0x00020000;                     // data_size=4B, wg_mask=0
  td_g1[1] = (int)(64u << 16);               // tensor_dim0 = 64
  td_g1[2] = (int)(16u << 16);               // tensor_dim1 = 16
  td_g1[3] = (int)(64u << 16);               // tile_dim0 = 64
  td_g1[4] = 16;                             // tile_dim1 = 16, tile_dim2 = 0
  td_g1[5] = 1024;                           // tensor_dim0_stride = 1024
  td_g1[6] = 0;
  td_g1[7] = 0;
#endif

  const int t0base = b * L_SEQ + lseg * 512;

  for (int lt = 0; lt < 512; lt += 16) {
    const int t0 = t0base + lt;

#if HAVE_TDM
    if (wave == 0) {  // one TDM issue per workgroup, overlapped with dash phase
      unsigned long long ga = (unsigned long long)(uintptr_t)(
          v + (size_t)t0 * D_MODEL + h * D_HEAD);
      td_g0[2] = (unsigned)ga;
      td_g0[3] = (unsigned)((ga >> 32) & 0x01FFFFFFull) | 0x80000000u;  // type=2
      __builtin_amdgcn_tensor_load_to_lds(td_g0, td_g1, td_g2, td_g3, td_g4, 0);
    }
#else
    {
      const int r = tid >> 4;
      const int c = (tid & 15) * 4;
      const float* src = v + (size_t)(t0 + r) * D_MODEL + h * D_HEAD + c;
      #pragma unroll
      for (int i = 0; i < 4; ++i) vS[r][c + i] = src[i];
    }
#endif

    // ---- phase A: dash_k for 16 tokens; wave -> feature tile `wave` ----
    v8f dacc = vzero8();
    {
      const int tok = t0 + l15;
      const float* prow = pos_proj + (size_t)tok * D_MODEL + h * D_HEAD;
      const float* srow = slope_proj + (size_t)tok * D_MODEL + h * D_HEAD;
      const int fc = wave * 16 + l15;
      #pragma unroll
      for (int ks = 0; ks < D_HEAD; ks += 4) {
        const int kk = ks + 2 * hi;
        v2f a, bf;
        a.x  = a_mul * prow[kk] + b_mul * srow[kk];
        a.y  = a_mul * prow[kk + 1] + b_mul * srow[kk + 1];
        bf.x = proj[fc * D_HEAD + kk];       // projT[k][feat]
        bf.y = proj[fc * D_HEAD + kk + 1];
        dacc = wmma4(a, bf, dacc);
      }
    }
    #pragma unroll
    for (int r = 0; r < 8; ++r) {
      const int m = r + 8 * hi;        // token local 0..15
      const int f = wave * 16 + l15;   // feature 0..127
      const float d = dacc[r];
      kT[f][m]          = RATIO_SCALE * __sinf(d);
      kT[f + M_FEAT][m] = RATIO_SCALE * __cosf(d);
    }
#if HAVE_TDM
    if (wave == 0) __builtin_amdgcn_s_wait_tensorcnt(0);
#endif
    __syncthreads();

    // ---- phase B: kv += k'^T (256 x 16) @ v (16 x 64) ----
    #pragma unroll
    for (int ks = 0; ks < 16; ks += 4) {
      v2f af[2], bf[4];
      #pragma unroll
      for (int mi = 0; mi < 2; ++mi) {
        const int fr = wave * 32 + mi * 16 + l15;
        af[mi].x = kT[fr][ks + 2 * hi];
        af[mi].y = kT[fr][ks + 2 * hi + 1];
      }
      #pragma unroll
      for (int nj = 0; nj < 4; ++nj) {
        const int col = nj * 16 + l15;
        bf[nj].x = vS[ks + 2 * hi][col];
        bf[nj].y = vS[ks + 2 * hi + 1][col];
      }
      #pragma unroll
      for (int mi = 0; mi < 2; ++mi)
        #pragma unroll
        for (int nj = 0; nj < 4; ++nj)
          acc[mi][nj] = wmma4(af[mi], bf[nj], acc[mi][nj]);
    }
    __syncthreads();  // protect kT / vS reuse next iteration
  }

  // merge partial kv
  const size_t base = ((size_t)(b * N_HEAD + h)) * 2 * M_FEAT * D_HEAD;
  #pragma unroll
  for (int mi = 0; mi < 2; ++mi)
    #pragma unroll
    for (int nj = 0; nj < 4; ++nj)
      #pragma unroll
      for (int r = 0; r < 8; ++r) {
        const int fr  = wave * 32 + mi * 16 + r + 8 * hi;
        const int col = nj * 16 + l15;
        atomicAdd(&kv_g[base + (size_t)fr * D_HEAD + col], acc[mi][nj][r]);
      }
}

// ---------------------------------------------------------------------------
// z kernel: per (b, h, 32-token tile): dash_q (WMMA) -> sin/cos -> q' in LDS
// -> z = q' @ kv (WMMA) -> scale by ||slope_proj||/L -> write z.
// ---------------------------------------------------------------------------
__global__ __launch_bounds__(256) void z_kernel(
    const float* __restrict__ pos_proj, const float* __restrict__ slope_proj,
    const float* __restrict__ kv_g, const float* __restrict__ proj,
    const float* __restrict__ scale_v, float* __restrict__ z) {
  const int wg = blockIdx.x;
  const int lt = wg & 127;
  const int h  = (wg >> 7) & 15;
  const int b  = wg >> 11;

  __shared__ float qp[32][2 * M_FEAT + 2];  // q' tile [tok][feat]
  __shared__ float normf[32];

  const int tid  = threadIdx.x;
  const int lane = tid & 31;
  const int wave = tid >> 5;
  const int l15  = lane & 15;
  const int hi   = lane >> 4;

  const float sc = scale_v[h];
  const int t0 = b * L_SEQ + lt * 32;

  // norm factor: ||slope_proj[tok, h, :]|| / L  (raw slope_proj, unscaled)
  if (tid < 32) {
    const float* srow = slope_proj + (size_t)(t0 + tid) * D_MODEL + h * D_HEAD;
    float s = 0.f;
    #pragma unroll 8
    for (int d = 0; d < D_HEAD; ++d) s += srow[d] * srow[d];
    normf[tid] = sqrtf(s) * (1.0f / (float)L_SEQ);
  }

  // ---- phase 1: dash_q (32 tokens x 128 features), wave -> feature tile ----
  v8f dacc[2] = {vzero8(), vzero8()};
  {
    const int fc = wave * 16 + l15;
    #pragma unroll
    for (int ks = 0; ks < D_HEAD; ks += 4) {
      const int kk = ks + 2 * hi;
      v2f bf;
      bf.x = proj[fc * D_HEAD + kk];
      bf.y = proj[fc * D_HEAD + kk + 1];
      #pragma unroll
      for (int mi = 0; mi < 2; ++mi) {
        const int tok = t0 + mi * 16 + l15;
        const float* prow = pos_proj + (size_t)tok * D_MODEL + h * D_HEAD;
        v2f a;
        a.x = DN_SCALE * sc * prow[kk];
        a.y = DN_SCALE * sc * prow[kk + 1];
        dacc[mi] = wmma4(a, bf, dacc[mi]);
      }
    }
  }
  #pragma unroll
  for (int mi = 0; mi < 2; ++mi)
    #pragma unroll
    for (int r = 0; r < 8; ++r) {
      const int m = mi * 16 + r + 8 * hi;  // token local 0..31
      const int f = wave * 16 + l15;
      const float d = dacc[mi][r];
      qp[m][f]          = RATIO_SCALE * __sinf(d);
      qp[m][f + M_FEAT] = RATIO_SCALE * __cosf(d);
    }
  __syncthreads();

  // ---- phase 2: z (32 x 64) = q' (32 x 256) @ kv (256 x 64) ----
  const float* kv = kv_g + ((size_t)(b * N_HEAD + h)) * 2 * M_FEAT * D_HEAD;
  const int mi = wave & 1;
  const int nj = wave >> 1;
  v8f zacc = vzero8();
  #pragma unroll 8
  for (int ks = 0; ks < 2 * M_FEAT; ks += 4) {
    v2f a, bf;
    const int row = mi * 16 + l15;
    a.x = qp[row][ks + 2 * hi];
    a.y = qp[row][ks + 2 * hi + 1];
    const int col = nj * 16 + l15;
    bf.x = kv[(size_t)(ks + 2 * hi) * D_HEAD + col];
    bf.y = kv[(size_t)(ks + 2 * hi + 1) * D_HEAD + col];
    zacc = wmma4(a, bf, zacc);
  }

  #pragma unroll
  for (int r = 0; r < 8; ++r) {
    const int m   = mi * 16 + r + 8 * hi;
    const int tok = t0 + m;
    const int col = nj * 16 + l15;
    z[(size_t)tok * D_MODEL + h * D_HEAD + col] = zacc[r] * normf[m];
  }
}

__global__ __launch_bounds__(256) void zero_kernel(float* __restrict__ p, int n) {
  const int i = blockIdx.x * 256 + threadIdx.x;
  if (i < n) p[i] = 0.f;
}

// ---------------------------------------------------------------------------
extern "C" void kernel_launch(void* const* d_in, const int* in_sizes, int n_in,
                              void* d_out, int out_size, void* d_ws, size_t ws_size,
                              hipStream_t stream) {
  const float* x       = (const float*)d_in[0];
  const float* pos_ft  = (const float*)d_in[1];
  const float* slopes  = (const float*)d_in[2];
  const float* W_v     = (const float*)d_in[3];
  const float* W_o     = (const float*)d_in[4];
  const float* W_pf    = (const float*)d_in[5];
  const float* scale_v = (const float*)d_in[6];
  const float* offs_v  = (const float*)d_in[7];
  const float* proj    = (const float*)d_in[8];
  float* out = (float*)d_out;

  const int M = 4 * L_SEQ;      // 16384 rows
  const int N = D_MODEL;        // 1024
  const int K = D_MODEL;        // 1024
  const size_t MAT = (size_t)M * N;

  float* ws         = (float*)d_ws;
  float* v_buf      = ws;                    // 64 MB; reused as z after kv
  float* pos_proj   = v_buf + MAT;           // 64 MB
  float* slope_proj = pos_proj + MAT;        // 64 MB
  float* kv_buf     = slope_proj + MAT;      // 4 MB (64 * 256 * 64)
  const int KV_ELEMS = 4 * N_HEAD * 2 * M_FEAT * D_HEAD;

  dim3 gemm_grid(M / BM, N / BN);

  // three input projections
  hipLaunchKernelGGL(gemm_f32_wmma, gemm_grid, dim3(256), 0, stream,
                     x, W_v, v_buf, M, N, K);
  hipLaunchKernelGGL(gemm_f32_wmma, gemm_grid, dim3(256), 0, stream,
                     pos_ft, W_pf, pos_proj, M, N, K);
  hipLaunchKernelGGL(gemm_f32_wmma, gemm_grid, dim3(256), 0, stream,
                     slopes, W_pf, slope_proj, M, N, K);

  // kv = sum_l k'^T v  (L split 8 ways, atomic merge)
  hipLaunchKernelGGL(zero_kernel, dim3((KV_ELEMS + 255) / 256), dim3(256), 0,
                     stream, kv_buf, KV_ELEMS);
  hipLaunchKernelGGL(kv_kernel, dim3(4 * N_HEAD * 8), dim3(256), 0, stream,
                     pos_proj, slope_proj, v_buf, proj, scale_v, offs_v, kv_buf);

  // z = (q' @ kv) * norm  — overwrites v_buf (v is dead now)
  hipLaunchKernelGGL(z_kernel, dim3(4 * N_HEAD * (L_SEQ / 32)), dim3(256), 0,
                     stream, pos_proj, slope_proj, kv_buf, proj, scale_v, v_buf);

  // out = z @ W_o
  hipLaunchKernelGGL(gemm_f32_wmma, gemm_grid, dim3(256), 0, stream,
                     v_buf, W_o, out, M, N, K);
}